// NeuralIF_46918222742294
// MI455X (gfx1250) — compile-verified
//
#include <hip/hip_runtime.h>
#include <math.h>

typedef float v2f __attribute__((ext_vector_type(2)));
typedef float v4f __attribute__((ext_vector_type(4)));
typedef float v8f __attribute__((ext_vector_type(8)));

#define LD 64

// ---------------- elementwise / setup kernels ----------------

__global__ void zero_kernel(float* __restrict__ p, long long n) {
    long long i = (long long)blockIdx.x * blockDim.x + threadIdx.x;
    if (i < n) p[i] = 0.0f;
}

__global__ void deg_cnt_kernel(const float* __restrict__ attr, const int* __restrict__ ecol,
                               float* __restrict__ deg, float* __restrict__ cnt, int E) {
    int e = blockIdx.x * blockDim.x + threadIdx.x;
    if (e < E) {
        int c = ecol[e];
        unsafeAtomicAdd(&deg[c], attr[e]);
        unsafeAtomicAdd(&cnt[c], 1.0f);
    }
}

__global__ void dinv_kernel(const float* __restrict__ deg, float* __restrict__ cnt,
                            float* __restrict__ dinv, int N) {
    int n = blockIdx.x * blockDim.x + threadIdx.x;
    if (n < N) {
        float d = deg[n];
        dinv[n] = (d > 0.0f) ? rsqrtf(fmaxf(d, 1e-30f)) : 0.0f;
        cnt[n] = fmaxf(cnt[n], 1.0f);
    }
}

__global__ void norm_kernel(const float* __restrict__ attr, const int* __restrict__ erow,
                            const int* __restrict__ ecol, const float* __restrict__ dinv,
                            float* __restrict__ nrm, int E) {
    int e = blockIdx.x * blockDim.x + threadIdx.x;
    if (e < E) nrm[e] = dinv[erow[e]] * attr[e] * dinv[ecol[e]];
}

__global__ void encoder_kernel(const float* __restrict__ x, const float* __restrict__ w,
                               const float* __restrict__ b, float* __restrict__ h, long long NL) {
    long long i = (long long)blockIdx.x * blockDim.x + threadIdx.x;
    if (i < NL) {
        int j = (int)(i & 63);
        h[i] = x[i >> 6] * w[j] + b[j];
    }
}

// ---------------- WMMA GEMM: hW[N,64] = h[N,64] @ W[64,64] ----------------
// One wave32 per 16-row tile; A staged in LDS (stride 66 to dodge bank conflicts).
// V_WMMA_F32_16X16X4_F32 layouts (ISA 7.12.2):
//   A: lane L<16 holds row L, K = kb+{0,1}; lane L>=16 holds row L-16, K = kb+{2,3}
//   B: lane L<16 holds col L, K = kb+{0,1}; lane L>=16 holds col L-16, K = kb+{2,3}
//   C: VGPR v, lanes 0-15 -> M=v, lanes 16-31 -> M=v+8; N = lane%16
__global__ __launch_bounds__(128) void gemm64_kernel(const float* __restrict__ h,
                                                     const float* __restrict__ W,
                                                     float* __restrict__ hW, int N) {
    __shared__ float As[4][16 * 66];
    const int lane = threadIdx.x & 31;
    const int wave = threadIdx.x >> 5;
    const int rb = (blockIdx.x * 4 + wave) * 16;
    float* At = As[wave];

    // stage 16x64 A tile (clamped rows keep EXEC all-ones)
    for (int m = 0; m < 16; ++m) {
        int r = rb + m; if (r > N - 1) r = N - 1; if (r < 0) r = 0;
        v2f t = *(const v2f*)(h + (long long)r * LD + lane * 2);
        At[m * 66 + lane * 2]     = t.x;
        At[m * 66 + lane * 2 + 1] = t.y;
    }
    __syncthreads();

    const int nlo = lane & 15;
    const int khalf = lane >> 4;
    for (int nt = 0; nt < 4; ++nt) {
        v8f acc = {0.f, 0.f, 0.f, 0.f, 0.f, 0.f, 0.f, 0.f};
        const int n = nt * 16 + nlo;
        for (int kb = 0; kb < 64; kb += 4) {
            const int ka = kb + 2 * khalf;
            v2f a; a.x = At[nlo * 66 + ka];  a.y = At[nlo * 66 + ka + 1];
            v2f b; b.x = W[ka * 64 + n];     b.y = W[(ka + 1) * 64 + n];
            acc = __builtin_amdgcn_wmma_f32_16x16x4_f32(false, a, false, b,
                                                        (short)0, acc, false, false);
        }
#pragma unroll
        for (int v = 0; v < 8; ++v) {
            int r = rb + v + 8 * khalf;
            if (r < N) hW[(long long)r * LD + n] = acc[v];
        }
    }
}

// ---------------- message scatter: agg[col] += hW[row] * norm ----------------
__global__ void scatter_kernel(const float* __restrict__ hW, const float* __restrict__ nrm,
                               const int* __restrict__ erow, const int* __restrict__ ecol,
                               float* __restrict__ agg, long long EL) {
    long long i = (long long)blockIdx.x * blockDim.x + threadIdx.x;
    if (i < EL) {
        int e = (int)(i >> 6);
        int j = (int)(i & 63);
        float m = hW[(long long)erow[e] * LD + j] * nrm[e];
        unsafeAtomicAdd(&agg[(long long)ecol[e] * LD + j], m);
    }
}

__global__ void finalize_kernel(const float* __restrict__ agg, const float* __restrict__ cnt,
                                const float* __restrict__ b, float* __restrict__ h, long long NL) {
    long long i = (long long)blockIdx.x * blockDim.x + threadIdx.x;
    if (i < NL) {
        float v = agg[i] / cnt[i >> 6] + b[i & 63];
        h[i] = fmaxf(v, 0.0f);
    }
}

// ---------------- edge decoder: relu([h_r,h_c] @ W1 + b1) @ w2 + b2 ----------------
// One wave per 16-edge tile; 16x128 gathered feature tile staged in LDS via b128 loads.
__global__ __launch_bounds__(128) void decoder_kernel(
        const float* __restrict__ h, const float* __restrict__ w1, const float* __restrict__ b1,
        const float* __restrict__ w2, const float* __restrict__ b2v,
        const int* __restrict__ erow, const int* __restrict__ ecol,
        float* __restrict__ out, int E) {
    __shared__ float As[4][16 * 132];
    const int lane = threadIdx.x & 31;
    const int wave = threadIdx.x >> 5;
    const int eb = (blockIdx.x * 4 + wave) * 16;
    float* At = As[wave];

    const int k0 = lane * 4;  // 32 lanes * 4 floats = 128-wide row
    for (int m = 0; m < 16; ++m) {
        int e = eb + m; if (e > E - 1) e = E - 1; if (e < 0) e = 0;
        int rr = erow[e], cc = ecol[e];
        const float* src = (k0 < 64) ? (h + (long long)rr * LD + k0)
                                     : (h + (long long)cc * LD + (k0 - 64));
        v4f t = *(const v4f*)src;
        *(v4f*)(At + m * 132 + k0) = t;
    }
    __syncthreads();

    const int nlo = lane & 15;
    const int khalf = lane >> 4;
    float p[8];
#pragma unroll
    for (int v = 0; v < 8; ++v) p[v] = 0.0f;

    for (int nt = 0; nt < 4; ++nt) {
        v8f acc = {0.f, 0.f, 0.f, 0.f, 0.f, 0.f, 0.f, 0.f};
        const int n = nt * 16 + nlo;
        for (int kb = 0; kb < 128; kb += 4) {
            const int ka = kb + 2 * khalf;
            v2f a; a.x = At[nlo * 132 + ka]; a.y = At[nlo * 132 + ka + 1];
            v2f b; b.x = w1[ka * 64 + n];    b.y = w1[(ka + 1) * 64 + n];
            acc = __builtin_amdgcn_wmma_f32_16x16x4_f32(false, a, false, b,
                                                        (short)0, acc, false, false);
        }
        const float bn = b1[n];
        const float wn = w2[n];
#pragma unroll
        for (int v = 0; v < 8; ++v) {
            float e1 = acc[v] + bn;
            e1 = (e1 > 0.0f) ? e1 : 0.0f;
            p[v] += e1 * wn;     // second decoder layer, fused
        }
    }

    // reduce over the 16 N-lanes of each half (wave32, width-16 xor shuffle)
#pragma unroll
    for (int v = 0; v < 8; ++v) {
#pragma unroll
        for (int off = 8; off >= 1; off >>= 1)
            p[v] += __shfl_xor(p[v], off, 16);
    }

    if (nlo == 0) {
        const float b2 = b2v[0];
#pragma unroll
        for (int v = 0; v < 8; ++v) {
            int e = eb + v + 8 * khalf;  // M index of C row
            if (e < E) {
                float val = p[v] + b2;
                if (erow[e] == ecol[e]) {
                    // stable softplus: max(x,0) + log1p(exp(-|x|))
                    val = fmaxf(val, 0.0f) + log1pf(expf(-fabsf(val)));
                }
                out[e] = val;
            }
        }
    }
}

// ---------------- host orchestration ----------------

extern "C" void kernel_launch(void* const* d_in, const int* in_sizes, int n_in,
                              void* d_out, int out_size, void* d_ws, size_t ws_size,
                              hipStream_t stream) {
    const float* x       = (const float*)d_in[0];
    const float* attr    = (const float*)d_in[1];
    const float* w_enc   = (const float*)d_in[2];
    const float* b_enc   = (const float*)d_in[3];
    const float* conv1_w = (const float*)d_in[4];
    const float* conv1_b = (const float*)d_in[5];
    const float* conv2_w = (const float*)d_in[6];
    const float* conv2_b = (const float*)d_in[7];
    const float* dec_w1  = (const float*)d_in[8];
    const float* dec_b1  = (const float*)d_in[9];
    const float* dec_w2  = (const float*)d_in[10];
    const float* dec_b2  = (const float*)d_in[11];
    const int*   erow    = (const int*)d_in[12];
    const int*   ecol    = (const int*)d_in[13];

    const int N = in_sizes[0];   // x is (N,1)
    const int E = in_sizes[1];   // edge_attr is (E,)
    const long long NL = (long long)N * 64;
    const long long EL = (long long)E * 64;

    char* ws = (char*)d_ws;
    size_t off = 0;
    auto take = [&](size_t count) -> float* {
        float* p = (float*)(ws + off);
        off += (count * sizeof(float) + 255) & ~(size_t)255;
        return p;
    };
    float* deg  = take(N);
    float* cnt  = take(N);
    float* dinv = take(N);
    float* nrm  = take(E);
    float* h    = take((size_t)NL);
    float* hW   = take((size_t)NL);
    float* agg  = take((size_t)NL);

    const int B = 256;
    zero_kernel<<<(int)((N + B - 1) / B), B, 0, stream>>>(deg, N);
    zero_kernel<<<(int)((N + B - 1) / B), B, 0, stream>>>(cnt, N);
    deg_cnt_kernel<<<(E + B - 1) / B, B, 0, stream>>>(attr, ecol, deg, cnt, E);
    dinv_kernel<<<(N + B - 1) / B, B, 0, stream>>>(deg, cnt, dinv, N);
    norm_kernel<<<(E + B - 1) / B, B, 0, stream>>>(attr, erow, ecol, dinv, nrm, E);
    encoder_kernel<<<(int)((NL + B - 1) / B), B, 0, stream>>>(x, w_enc, b_enc, h, NL);

    const int gemm_grid = (N + 63) / 64;  // 4 waves/block * 16 rows/wave
    for (int l = 0; l < 6; ++l) {
        const float* W  = ((l & 1) ? conv2_w : conv1_w) + (size_t)(l >> 1) * 64 * 64;
        const float* bb = ((l & 1) ? conv2_b : conv1_b) + (size_t)(l >> 1) * 64;
        gemm64_kernel<<<gemm_grid, 128, 0, stream>>>(h, W, hW, N);
        zero_kernel<<<(int)((NL + B - 1) / B), B, 0, stream>>>(agg, NL);
        scatter_kernel<<<(int)((EL + B - 1) / B), B, 0, stream>>>(hW, nrm, erow, ecol, agg, EL);
        finalize_kernel<<<(int)((NL + B - 1) / B), B, 0, stream>>>(agg, cnt, bb, h, NL);
    }

    const int dec_grid = (E + 63) / 64;   // 4 waves/block * 16 edges/wave
    decoder_kernel<<<dec_grid, 128, 0, stream>>>(h, dec_w1, dec_b1, dec_w2, dec_b2,
                                                 erow, ecol, (float*)d_out, E);
}